// MultiDepthLimitedMSELoss_1941325218285
// MI455X (gfx1250) — compile-verified
//
#include <hip/hip_runtime.h>

// CDNA5 / gfx1250: wave32. Ext-vector types for clean codegen.
typedef __attribute__((ext_vector_type(4))) float f4;
typedef __attribute__((ext_vector_type(2))) float v2f;
typedef __attribute__((ext_vector_type(8))) float v8f;

#define WAVE 32
#define BLOCK 256
#define NBLOCKS 4096
#define BIGF 3.402823466e+38f

// ---------------------------------------------------------------------------
// Kernel 1: streaming greedy-match + per-block partial loss sum.
// Bandwidth-bound: 64B/row streamed once -> nontemporal b128 loads.
// All matching state lives in registers (D=8, fully unrolled select chains).
// ---------------------------------------------------------------------------
__global__ void __launch_bounds__(BLOCK)
mdl_mse_partial_kernel(const float* __restrict__ outputs,
                       const float* __restrict__ targets,
                       float* __restrict__ partials,
                       int nrows) {
  const int tid    = blockIdx.x * BLOCK + threadIdx.x;
  const int stride = gridDim.x * BLOCK;

  float acc = 0.0f;

  for (int r = tid; r < nrows; r += stride) {
    const f4* op = (const f4*)(outputs + (size_t)r * 8);
    const f4* tp = (const f4*)(targets + (size_t)r * 8);
    // Streaming data (256MB > 192MB L2): non-temporal hint.
    f4 o0 = __builtin_nontemporal_load(op);
    f4 o1 = __builtin_nontemporal_load(op + 1);
    f4 t0 = __builtin_nontemporal_load(tp);
    f4 t1 = __builtin_nontemporal_load(tp + 1);

    float o[8], t[8], cur[8];
#pragma unroll
    for (int j = 0; j < 4; ++j) { o[j] = o0[j]; o[j + 4] = o1[j]; }
#pragma unroll
    for (int j = 0; j < 4; ++j) { t[j] = t0[j]; t[j + 4] = t1[j]; }
#pragma unroll
    for (int j = 0; j < 8; ++j) cur[j] = o[j];

    float loss = 0.0f;
#pragma unroll
    for (int i = 0; i < 8; ++i) {
      const float ti = t[i];
      // argmin_j |cur[j] - ti|, first-index wins ties (strict <), like jnp.argmin
      float best = fabsf(cur[0] - ti);
      int   bid  = 0;
#pragma unroll
      for (int j = 1; j < 8; ++j) {
        float d = fabsf(cur[j] - ti);
        bool lt = d < best;
        best = lt ? d : best;
        bid  = lt ? j : bid;
      }
      // gather matched output (register select chain) and knock out the slot
      float osh = o[0];
#pragma unroll
      for (int j = 1; j < 8; ++j) osh = (bid == j) ? o[j] : osh;
#pragma unroll
      for (int j = 0; j < 8; ++j) cur[j] = (bid == j) ? BIGF : cur[j];
      // seltargets: if target == IGNORE(0), error term is exactly 0
      float diff = osh - ti;
      loss += (ti != 0.0f) ? diff * diff : 0.0f;
    }
    acc += loss;
  }

  // Deterministic wave32 reduction (fixed xor tree).
#pragma unroll
  for (int off = 16; off > 0; off >>= 1)
    acc += __shfl_xor(acc, off, WAVE);

  __shared__ float wsum[BLOCK / WAVE];
  const int wid = threadIdx.x >> 5;
  if ((threadIdx.x & (WAVE - 1)) == 0) wsum[wid] = acc;
  __syncthreads();
  if (threadIdx.x == 0) {
    float s = 0.0f;
#pragma unroll
    for (int w = 0; w < BLOCK / WAVE; ++w) s += wsum[w];
    partials[blockIdx.x] = s;  // fixed order -> deterministic
  }
}

// ---------------------------------------------------------------------------
// Kernel 2: single-block finisher. 4096 partials -> 256 thread sums (fixed
// stride order) -> exact f32 reduction through the matrix pipe:
// four V_WMMA_F32_16X16X4_F32 with A = data tiles, B = ones, accumulating C.
// D[m][n] = sum_k A[m][k]  (exact f32 FMA, fixed dataflow, deterministic).
// A 16x4 f32 layout: lane l -> row m = l&15, k = 2*(l>>4) + vgpr_index.
// D 16x16 f32 layout: vgpr v -> row v (lanes 0-15) / row v+8 (lanes 16-31).
// ---------------------------------------------------------------------------
__global__ void __launch_bounds__(BLOCK)
mdl_mse_finish_kernel(const float* __restrict__ partials, int n,
                      float* __restrict__ out, float scale) {
  __shared__ float red[BLOCK];
  float s = 0.0f;
  for (int i = threadIdx.x; i < n; i += BLOCK) s += partials[i];
  red[threadIdx.x] = s;
  __syncthreads();

  if (threadIdx.x < WAVE) {           // wave 0 only: EXEC all-ones for WMMA
    const int l     = threadIdx.x;
    const int m     = l & 15;
    const int khalf = l >> 4;         // 0 -> K={0,1}, 1 -> K={2,3}
    v8f acc = {};
    v2f ones = {1.0f, 1.0f};
#pragma unroll
    for (int c = 0; c < 4; ++c) {     // 4 tiles of 64 cover red[0..255] once
      const int base = c * 64 + m * 4 + khalf * 2;
      v2f a;
      a[0] = red[base];
      a[1] = red[base + 1];
      acc = __builtin_amdgcn_wmma_f32_16x16x4_f32(
          /*neg_a=*/false, a, /*neg_b=*/false, ones,
          /*c_mod=*/(short)0, acc, /*reuse_a=*/false, /*reuse_b=*/false);
    }
    // lane<16 holds rowsums 0..7 (one per C vgpr), lane>=16 holds 8..15
    float tot = acc[0] + acc[1] + acc[2] + acc[3] +
                acc[4] + acc[5] + acc[6] + acc[7];
    tot += __shfl_xor(tot, 16, WAVE);
    if (l == 0) out[0] = tot * scale;
  }
}

// ---------------------------------------------------------------------------
extern "C" void kernel_launch(void* const* d_in, const int* in_sizes, int n_in,
                              void* d_out, int out_size, void* d_ws, size_t ws_size,
                              hipStream_t stream) {
  const float* outputs = (const float*)d_in[0];
  const float* targets = (const float*)d_in[1];
  float* out = (float*)d_out;
  float* partials = (float*)d_ws;          // NBLOCKS floats = 16 KB scratch

  const int nrows = in_sizes[0] / 8;       // N (D = 8)
  const float scale = 1.0f / ((float)nrows * 8.0f);  // mean over N*D

  mdl_mse_partial_kernel<<<NBLOCKS, BLOCK, 0, stream>>>(outputs, targets,
                                                        partials, nrows);
  mdl_mse_finish_kernel<<<1, BLOCK, 0, stream>>>(partials, NBLOCKS, out, scale);
}